// AdaptPooling_42984032698889
// MI455X (gfx1250) — compile-verified
//
#include <hip/hip_runtime.h>
#include <math.h>
#include <stdint.h>

// Problem constants (match reference)
#define N_NODES 200000
#define DIM     256
#define NH      4
#define NC      64
#define HC      (NH*NC)     // 256
#define NB      2000
#define NE      3200000
#define NP      3

typedef float v2f __attribute__((ext_vector_type(2)));
typedef float v8f __attribute__((ext_vector_type(8)));

// ---- gfx1250 async global->LDS copy (ASYNCcnt-tracked), with safe fallback --
#if defined(__AMDGCN__) && __has_builtin(__builtin_amdgcn_global_load_async_to_lds_b128)
#define HAVE_ASYNC_LDS 1
#else
#define HAVE_ASYNC_LDS 0
#endif

#if HAVE_ASYNC_LDS
// Builtin signature (from hipcc diagnostic): pointee is vector_size(16) int.
typedef int v4i_vs __attribute__((vector_size(16)));
typedef __attribute__((address_space(1))) v4i_vs* as1_v4i_ptr;
typedef __attribute__((address_space(3))) v4i_vs* as3_v4i_ptr;
#endif

__device__ __forceinline__ void async_copy_b128(const float* g, float* l) {
#if HAVE_ASYNC_LDS
  // Flat pointers: global flat address == AS1 address; flat-to-LDS uses the
  // low 32 bits as the LDS byte address (ISA 10.2 aperture rules). Go through
  // integers to avoid illegal flat->AS pointer casts.
  __builtin_amdgcn_global_load_async_to_lds_b128(
      (as1_v4i_ptr)(uintptr_t)g,
      (as3_v4i_ptr)(uint32_t)(uintptr_t)l,
      /*imm offset=*/0, /*cpol=*/0);
#else
  *(float4*)l = *(const float4*)g;
#endif
}

__device__ __forceinline__ void async_copy_wait() {
#if HAVE_ASYNC_LDS
#if __has_builtin(__builtin_amdgcn_s_wait_asynccnt)
  __builtin_amdgcn_s_wait_asynccnt(0);
#else
  asm volatile("s_wait_asynccnt 0x0" ::: "memory");
#endif
#endif
}

// ---------------------------------------------------------------------------
// K0: fold seed into W_key:  w_alpha[d][h] = (1/sqrt(C)) * sum_c Wk[d, h*C+c]*seed[h,c]
//     b_alpha[h]            = (1/sqrt(C)) * sum_c bk[h*C+c]*seed[h,c]
// ---------------------------------------------------------------------------
__global__ void k0_fold_seed(const float* __restrict__ Wk,
                             const float* __restrict__ bk,
                             const float* __restrict__ seed,
                             float* __restrict__ w_alpha,
                             float* __restrict__ b_alpha) {
  const float inv = 0.125f; // 1/sqrt(64)
  int d = blockIdx.x * blockDim.x + threadIdx.x;
  if (d < DIM) {
    for (int h = 0; h < NH; ++h) {
      const float* wrow = Wk + (size_t)d * HC + h * NC;
      const float* srow = seed + h * NC;
      float s = 0.f;
      for (int c = 0; c < NC; ++c) s += wrow[c] * srow[c];
      w_alpha[d * NH + h] = s * inv;
    }
  }
  if (blockIdx.x == 0 && threadIdx.x < NH) {
    int h = threadIdx.x;
    float s = 0.f;
    for (int c = 0; c < NC; ++c) s += bk[h * NC + c] * seed[h * NC + c];
    b_alpha[h] = s * inv;
  }
}

// ---------------------------------------------------------------------------
// K_segptr: CSR-style segment offsets from sorted batch ids.
// ---------------------------------------------------------------------------
__global__ void k_segptr(const int* __restrict__ batch, int* __restrict__ seg_ptr) {
  int n = blockIdx.x * blockDim.x + threadIdx.x;
  if (n >= N_NODES) return;
  int b = batch[n];
  if (n == 0) {
    for (int bb = 0; bb <= b; ++bb) seg_ptr[bb] = 0;
  } else {
    int pb = batch[n - 1];
    for (int bb = pb + 1; bb <= b; ++bb) seg_ptr[bb] = n;
  }
  if (n == N_NODES - 1) {
    for (int bb = b + 1; bb <= NB; ++bb) seg_ptr[bb] = N_NODES;
  }
}

// ---------------------------------------------------------------------------
// K1: alpha[n,h] = x[n,:] . w_alpha[:,h] + b_alpha[h]   (one wave32 per node)
// ---------------------------------------------------------------------------
__global__ void k1_alpha(const float* __restrict__ x,
                         const float* __restrict__ w_alpha,
                         const float* __restrict__ b_alpha,
                         float* __restrict__ alpha) {
  int wave = (int)((blockIdx.x * (size_t)blockDim.x + threadIdx.x) >> 5);
  int lane = threadIdx.x & 31;
  if (wave >= N_NODES) return;
  const float4* xr = (const float4*)(x + (size_t)wave * DIM);
  float a0 = 0.f, a1 = 0.f, a2 = 0.f, a3 = 0.f;
  for (int q = 0; q < 2; ++q) {
    int f4 = lane * 2 + q;
    float4 xv = xr[f4];
    const float4* wv = (const float4*)(w_alpha + f4 * 4 * NH);
    float4 w0 = wv[0], w1 = wv[1], w2 = wv[2], w3 = wv[3];
    a0 += xv.x * w0.x + xv.y * w1.x + xv.z * w2.x + xv.w * w3.x;
    a1 += xv.x * w0.y + xv.y * w1.y + xv.z * w2.y + xv.w * w3.y;
    a2 += xv.x * w0.z + xv.y * w1.z + xv.z * w2.z + xv.w * w3.z;
    a3 += xv.x * w0.w + xv.y * w1.w + xv.z * w2.w + xv.w * w3.w;
  }
  for (int off = 16; off > 0; off >>= 1) {
    a0 += __shfl_xor(a0, off, 32);
    a1 += __shfl_xor(a1, off, 32);
    a2 += __shfl_xor(a2, off, 32);
    a3 += __shfl_xor(a3, off, 32);
  }
  if (lane == 0) {
    float4 o = make_float4(a0 + b_alpha[0], a1 + b_alpha[1],
                           a2 + b_alpha[2], a3 + b_alpha[3]);
    ((float4*)(alpha + (size_t)wave * NH))[0] = o;
  }
}

// ---------------------------------------------------------------------------
// K2: one workgroup (256 thr) per segment (sorted batch => no global atomics).
//   1) block max of alpha per head   2) block sum of exp per head
//   3) thread t owns column d=t; stream x rows (prefetch ahead), per-node
//      softmax weights staged in LDS, accumulate s_w[b,h,d]
//   4) position mean
// ---------------------------------------------------------------------------
__global__ void k2_segment(const float* __restrict__ x,
                           const float* __restrict__ alpha,
                           const float* __restrict__ position,
                           const int* __restrict__ seg_ptr,
                           float* __restrict__ s_w,
                           float* __restrict__ new_pos) {
  int b = blockIdx.x;
  int t = threadIdx.x;
  int start = seg_ptr[b], end = seg_ptr[b + 1];
  int cnt = end - start;

  __shared__ float4 red[256];
  __shared__ float4 wsh[64];
  __shared__ float  smax[4], sinv[4];
  __shared__ float  spos[3];

  // phase 1: per-head max
  float4 m = make_float4(-__builtin_inff(), -__builtin_inff(),
                         -__builtin_inff(), -__builtin_inff());
  for (int n = start + t; n < end; n += 256) {
    float4 a = ((const float4*)alpha)[n];
    m.x = fmaxf(m.x, a.x); m.y = fmaxf(m.y, a.y);
    m.z = fmaxf(m.z, a.z); m.w = fmaxf(m.w, a.w);
  }
  red[t] = m; __syncthreads();
  for (int s = 128; s > 0; s >>= 1) {
    if (t < s) {
      float4 o = red[t + s];
      red[t].x = fmaxf(red[t].x, o.x); red[t].y = fmaxf(red[t].y, o.y);
      red[t].z = fmaxf(red[t].z, o.z); red[t].w = fmaxf(red[t].w, o.w);
    }
    __syncthreads();
  }
  if (t == 0) { float4 mm = red[0]; smax[0]=mm.x; smax[1]=mm.y; smax[2]=mm.z; smax[3]=mm.w; }
  __syncthreads();
  float4 sm = make_float4(smax[0], smax[1], smax[2], smax[3]);

  // phase 2: per-head sum of exp
  float4 sa = make_float4(0.f, 0.f, 0.f, 0.f);
  for (int n = start + t; n < end; n += 256) {
    float4 a = ((const float4*)alpha)[n];
    sa.x += __expf(a.x - sm.x); sa.y += __expf(a.y - sm.y);
    sa.z += __expf(a.z - sm.z); sa.w += __expf(a.w - sm.w);
  }
  red[t] = sa; __syncthreads();
  for (int s = 128; s > 0; s >>= 1) {
    if (t < s) {
      float4 o = red[t + s];
      red[t].x += o.x; red[t].y += o.y; red[t].z += o.z; red[t].w += o.w;
    }
    __syncthreads();
  }
  if (t == 0) {
    float4 dd = red[0];
    sinv[0] = 1.f / fmaxf(dd.x, 1e-16f); sinv[1] = 1.f / fmaxf(dd.y, 1e-16f);
    sinv[2] = 1.f / fmaxf(dd.z, 1e-16f); sinv[3] = 1.f / fmaxf(dd.w, 1e-16f);
    spos[0] = 0.f; spos[1] = 0.f; spos[2] = 0.f;
  }
  __syncthreads();

  // phase 3: weighted segment-sum of x
  int d = t;
  float acc0 = 0.f, acc1 = 0.f, acc2 = 0.f, acc3 = 0.f;
  for (int n0 = start; n0 < end; n0 += 64) {
    int nn = min(64, end - n0);
    __syncthreads();
    if (t < nn * 4) {
      int ni = t >> 2, h = t & 3;
      float a = alpha[(size_t)(n0 + ni) * NH + h];
      ((float*)wsh)[t] = __expf(a - smax[h]) * sinv[h];
    }
    __syncthreads();
    for (int j = 0; j < nn; ++j) {
      float xv = x[(size_t)(n0 + j) * DIM + d];          // coalesced 1KB row
      if (n0 + j + 8 < end)
        __builtin_prefetch(x + (size_t)(n0 + j + 8) * DIM + d, 0, 1);
      float4 w = wsh[j];                                  // LDS b128 broadcast
      acc0 += w.x * xv; acc1 += w.y * xv; acc2 += w.z * xv; acc3 += w.w * xv;
    }
  }
  float* srow = s_w + ((size_t)b * NH) * DIM + d;
  srow[0 * DIM] = acc0; srow[1 * DIM] = acc1;
  srow[2 * DIM] = acc2; srow[3 * DIM] = acc3;

  // phase 4: position mean
  if (t < 96) {
    int p = t % 3, i0 = t / 3;
    float s = 0.f;
    for (int n = start + i0; n < end; n += 32) s += position[(size_t)n * NP + p];
    atomicAdd(&spos[p], s);
  }
  __syncthreads();
  if (t < 3) new_pos[(size_t)b * NP + t] = spos[t] / (float)max(cnt, 1);
}

// ---------------------------------------------------------------------------
// K5: out = s_w @ W_value (per head) + b_value, via v_wmma_f32_16x16x4_f32.
// Block = 8 waves; block stages its 256x16 W_value tile (16KB) into LDS once
// with GLOBAL_LOAD_ASYNC_TO_LDS_B128 (ASYNCcnt), then each wave computes one
// 16x16 output tile: A streamed from global (b64/lane), B fed by ds_load
// broadcasts from LDS. K-loop = 64 WMMAs.
// A layout (16x4 f32): lanes 0-15 hold M=lane, K={0,1}; lanes 16-31 K={2,3}.
// B layout (4x16 f32): lanes hold N=lane&15, same K split.
// C/D: 8 VGPRs, VGPR r -> M=r (lanes 0-15) / M=r+8 (lanes 16-31), N=lane&15.
// ---------------------------------------------------------------------------
#define K5_WAVES 8
#define MTILES   (NB / 16)   // 125

__global__ void k5_gemm_wmma(const float* __restrict__ s_w,
                             const float* __restrict__ Wv,
                             const float* __restrict__ bv,
                             const int* __restrict__ seg_ptr,
                             float* __restrict__ out) {
  __shared__ float btile[DIM * 16];            // [k][16], 16 KB

  int t  = threadIdx.x;                        // 256 threads
  int wv = t >> 5;
  int lane = t & 31;
  int n0 = blockIdx.y * 16;                    // column tile within HC=256
  int h  = n0 >> 6;                            // head owning this tile

  // Stage B tile: 256 rows x 16 cols = 1024 b128 words, 4 per thread.
  {
    const float* src = Wv + n0;                // row k at src + k*HC
    for (int w = t; w < DIM * 4; w += 256) {
      int k  = w >> 2;
      int c4 = (w & 3) * 4;
      async_copy_b128(src + (size_t)k * HC + c4, btile + k * 16 + c4);
    }
  }
  async_copy_wait();
  __syncthreads();

  int m0 = (blockIdx.x * K5_WAVES + wv) * 16;  // row tile
  if (m0 < NB) {
    int half = lane >> 4, lr = lane & 15;
    v8f c = {0.f, 0.f, 0.f, 0.f, 0.f, 0.f, 0.f, 0.f};
    const float* arow = s_w + ((size_t)(m0 + lr) * NH + h) * DIM;

    for (int k0 = 0; k0 < DIM; k0 += 4) {
      int ka = k0 + 2 * half;
      v2f a;  a.x  = arow[ka];              a.y  = arow[ka + 1];
      v2f bb; bb.x = btile[ka * 16 + lr];   bb.y = btile[(ka + 1) * 16 + lr];
      c = __builtin_amdgcn_wmma_f32_16x16x4_f32(
          /*neg_a=*/false, a, /*neg_b=*/false, bb,
          /*c_mod=*/(short)0, c, /*reuse_a=*/false, /*reuse_b=*/false);
    }

    int col = n0 + lr;
    float bias = bv[col];
    for (int r = 0; r < 8; ++r) {
      int row = m0 + r + half * 8;
      int cr  = seg_ptr[row + 1] - seg_ptr[row];
      out[(size_t)row * HC + col] = c[r] + (cr > 0 ? bias : 0.f);  // empty seg -> 0
    }
  }
}

// ---------------------------------------------------------------------------
// K6: edge remap + symmetrize: e = batch[raw]; out rows [e0|e1] and [e1|e0].
// ---------------------------------------------------------------------------
__global__ void k6_edges(const int* __restrict__ raw,
                         const int* __restrict__ batch,
                         float* __restrict__ eout) {
  int i = blockIdx.x * blockDim.x + threadIdx.x;
  if (i >= NE) return;
  int v0 = batch[raw[i]];
  int v1 = batch[raw[(size_t)NE + i]];
  const size_t E2 = 2 * (size_t)NE;
  eout[i]                   = (float)v0;
  eout[(size_t)NE + i]      = (float)v1;
  eout[E2 + i]              = (float)v1;
  eout[E2 + (size_t)NE + i] = (float)v0;
}

// ---------------------------------------------------------------------------
extern "C" void kernel_launch(void* const* d_in, const int* in_sizes, int n_in,
                              void* d_out, int out_size, void* d_ws, size_t ws_size,
                              hipStream_t stream) {
  (void)in_sizes; (void)n_in; (void)out_size; (void)ws_size;

  const float* x        = (const float*)d_in[0];
  const float* position = (const float*)d_in[1];
  const int*   batch    = (const int*)d_in[2];
  const int*   raw_edge = (const int*)d_in[3];
  // d_in[4] = num_segments (compile-time NB)
  const float* Wk   = (const float*)d_in[5];
  const float* bk   = (const float*)d_in[6];
  const float* Wv   = (const float*)d_in[7];
  const float* bv   = (const float*)d_in[8];
  const float* seed = (const float*)d_in[9];

  float* out      = (float*)d_out;                    // [NB, HC]
  float* edge_out = out + (size_t)NB * HC;            // [2, 2E]
  float* pos_out  = edge_out + 4 * (size_t)NE;        // [NB, NP]

  char* ws = (char*)d_ws;
  float* w_alpha = (float*)ws;  ws += DIM * NH * sizeof(float);          // 4 KB
  float* b_alpha = (float*)ws;  ws += 16 * sizeof(float);                // pad
  int*   seg_ptr = (int*)ws;    ws += ((NB + 1 + 15) / 16) * 16 * sizeof(int);
  float* alpha   = (float*)ws;  ws += (size_t)N_NODES * NH * sizeof(float); // 3.2 MB
  float* s_w     = (float*)ws;  // NB*NH*DIM floats = 8.2 MB

  k0_fold_seed<<<1, 256, 0, stream>>>(Wk, bk, seed, w_alpha, b_alpha);
  k_segptr<<<(N_NODES + 255) / 256, 256, 0, stream>>>(batch, seg_ptr);
  k1_alpha<<<(N_NODES + 7) / 8, 256, 0, stream>>>(x, w_alpha, b_alpha, alpha);
  k2_segment<<<NB, 256, 0, stream>>>(x, alpha, position, seg_ptr, s_w, pos_out);
  k5_gemm_wmma<<<dim3((MTILES + K5_WAVES - 1) / K5_WAVES, HC / 16), 256, 0, stream>>>(
      s_w, Wv, bv, seg_ptr, out);
  k6_edges<<<(NE + 255) / 256, 256, 0, stream>>>(raw_edge, batch, edge_out);
}